// SpatialModule_28776280883755
// MI455X (gfx1250) — compile-verified
//
#include <hip/hip_runtime.h>
#include <stdint.h>

#define AS1 __attribute__((address_space(1)))
#define AS3 __attribute__((address_space(3)))

typedef int   v4i __attribute__((vector_size(16)));
typedef float f4  __attribute__((ext_vector_type(4)));

#define TILE_H     16
#define LDS_STRIDE 528   // 512 data + 4 left pad + 12 right pad; float4-aligned rows
#define MAX_ROWS   24    // TILE_H + 2*d_max (d_max = 4)

// ---- gfx1250 async global->LDS copy (ASYNCcnt path, builtin confirmed) ------
__device__ __forceinline__ void async_copy_b128(float* lds_dst, const float* gsrc) {
    __builtin_amdgcn_global_load_async_to_lds_b128(
        (AS1 v4i*)(AS1 void*)gsrc,
        (AS3 v4i*)(AS3 void*)lds_dst, 0, 0);
}

__device__ __forceinline__ void wait_asynccnt0() {
#if __has_builtin(__builtin_amdgcn_s_wait_asynccnt)
    __builtin_amdgcn_s_wait_asynccnt(0);
#else
    asm volatile("s_wait_asynccnt 0x0" ::: "memory");
#endif
}

// ---- depthwise 3x3 for one dilation group, fully specialized on D -----------
template <int D>
__device__ __forceinline__ void run_group(
    float* lds,
    const float* __restrict__ x,
    const float* __restrict__ wsel,
    const float* __restrict__ bsel,
    float* __restrict__ out,
    int b, int c, int tileh, int tid)
{
    constexpr int G = (D == 4) ? 2 : (D == 2) ? 1 : 0;
    constexpr int R = TILE_H + 2 * D;            // 18 / 20 / 24 rows staged
    const int h0 = tileh * TILE_H;
    const int ci = G * 64 + c;                   // input channel

    // --- zero halo columns and out-of-bounds rows (ds_store, disjoint from async) ---
#pragma unroll
    for (int lr = 0; lr < R; ++lr) {
        const int gh = h0 - D + lr;
        if ((unsigned)gh >= 512u) {
            for (int i = tid; i < LDS_STRIDE; i += 256)
                lds[lr * LDS_STRIDE + i] = 0.0f;
        } else if (tid < 16) {
            const int cidx = (tid < 4) ? tid : (512 + tid);   // cols 0..3, 516..527
            lds[lr * LDS_STRIDE + cidx] = 0.0f;
        }
    }

    // --- async-stage valid rows into LDS interior (R*128 % 256 == 0: no tail) ---
    const float* xin = x + ((size_t)(b * 192 + ci) * 512) * 512;
    constexpr int N4 = R * 128;
#pragma unroll
    for (int i0 = 0; i0 < N4; i0 += 256) {
        const int i  = i0 + tid;
        const int lr = i >> 7;
        const int q  = i & 127;
        const int gh = h0 - D + lr;
        if ((unsigned)gh < 512u)
            async_copy_b128(&lds[lr * LDS_STRIDE + 4 + q * 4],
                            xin + (size_t)gh * 512 + q * 4);
    }
    wait_asynccnt0();      // drain this wave's ASYNCcnt
    __syncthreads();       // every wave drained before signaling -> cross-wave visible

    // --- weights/bias (block-uniform -> scalar loads) ---
    float wk[9];
#pragma unroll
    for (int k = 0; k < 9; ++k) wk[k] = wsel[c * 9 + k];
    const float bias = bsel[c];

    // output channel interleave: out[:, c*3+G]
    float* obase = out + (((size_t)(b * 192 + (c * 3 + G))) * 512 + h0) * 512;

#pragma unroll
    for (int pass = 0; pass < 8; ++pass) {
        const int row = pass * 2 + (tid >> 7);   // 0..15
        const int col = (tid & 127) << 2;        // 0,4,...,508
        float a0 = bias, a1 = bias, a2 = bias, a3 = bias;
#pragma unroll
        for (int ky = 0; ky < 3; ++ky) {
            // span of input cols [col-D, col+3+D] for this row tap, loaded once
            const float* p = &lds[(row + ky * D) * LDS_STRIDE + 4 + col - D];
            float s[4 + 2 * D];
#pragma unroll
            for (int j = 0; j < 4 + 2 * D; ++j) s[j] = p[j];
#pragma unroll
            for (int kx = 0; kx < 3; ++kx) {
                const float w = wk[ky * 3 + kx];
                a0 = fmaf(w, s[kx * D + 0], a0);
                a1 = fmaf(w, s[kx * D + 1], a1);
                a2 = fmaf(w, s[kx * D + 2], a2);
                a3 = fmaf(w, s[kx * D + 3], a3);
            }
        }
        f4 v = {a0, a1, a2, a3};
        // write-once output: non-temporal b128 store, keep L2 for input halos
        __builtin_nontemporal_store(v, (f4*)(obase + (size_t)row * 512 + col));
    }
}

__global__ __launch_bounds__(256) void dwconv3_interleave_kernel(
    const float* __restrict__ x,
    const float* __restrict__ w1, const float* __restrict__ b1,
    const float* __restrict__ w2, const float* __restrict__ b2,
    const float* __restrict__ w3, const float* __restrict__ b3,
    float* __restrict__ out)
{
    __shared__ float lds[MAX_ROWS * LDS_STRIDE];

    const int tid = threadIdx.x;
    int t = blockIdx.x;
    const int tileh = t & 31;  t >>= 5;          // 512/16 = 32 row-tiles
    const int ci    = t % 192;                   // input channel 0..191
    const int b     = t / 192;                   // batch 0..3
    const int g     = ci >> 6;                   // dilation group (block-uniform)
    const int c     = ci & 63;

    if (g == 0)      run_group<1>(lds, x, w1, b1, out, b, c, tileh, tid);
    else if (g == 1) run_group<2>(lds, x, w2, b2, out, b, c, tileh, tid);
    else             run_group<4>(lds, x, w3, b3, out, b, c, tileh, tid);
}

extern "C" void kernel_launch(void* const* d_in, const int* in_sizes, int n_in,
                              void* d_out, int out_size, void* d_ws, size_t ws_size,
                              hipStream_t stream) {
    const float* x  = (const float*)d_in[0];
    const float* w1 = (const float*)d_in[1];
    const float* b1 = (const float*)d_in[2];
    const float* w2 = (const float*)d_in[3];
    const float* b2 = (const float*)d_in[4];
    const float* w3 = (const float*)d_in[5];
    const float* b3 = (const float*)d_in[6];
    float* out = (float*)d_out;

    // B=4, 192 channels, 32 row-tiles of 16 rows each
    dim3 grid(4 * 192 * 32);
    dwconv3_interleave_kernel<<<grid, 256, 0, stream>>>(x, w1, b1, w2, b2, w3, b3, out);
}